// gnn_backbone_35880156791097
// MI455X (gfx1250) — compile-verified
//
#include <hip/hip_runtime.h>

typedef float v2f __attribute__((ext_vector_type(2)));
typedef float v8f __attribute__((ext_vector_type(8)));

#define FDIM 64

// ---------------------------------------------------------------------------
// Zero the hop buffers (float4 stores)
// ---------------------------------------------------------------------------
__global__ __launch_bounds__(256) void zero_kernel(float4* __restrict__ p, long n4) {
    long i = (long)blockIdx.x * 256 + threadIdx.x;
    if (i < n4) p[i] = make_float4(0.f, 0.f, 0.f, 0.f);
}

// ---------------------------------------------------------------------------
// One sparse propagation pass: hout[dst] += ew * hin[src]
// One wave32 per edge; each lane handles a float2 (features 2*lane, 2*lane+1).
// Gather of hin row is a coalesced 256B read (hits L2: feature matrix is
// 25.6MB << 192MB L2). Scatter via global_atomic_add_f32.
// ---------------------------------------------------------------------------
__global__ __launch_bounds__(256) void propagate_kernel(
    const float* __restrict__ hin, float* __restrict__ hout,
    const int* __restrict__ src, const int* __restrict__ dst,
    const float* __restrict__ ew, int E)
{
    int e = blockIdx.x * 8 + (threadIdx.x >> 5);
    if (e >= E) return;
    int lane = threadIdx.x & 31;
    int s = src[e];
    int d = dst[e];
    float w = ew[e];
    const v2f* ps = (const v2f*)(hin + (size_t)s * FDIM);
    v2f v = ps[lane];
    float* pd = hout + (size_t)d * FDIM + lane * 2;
    atomicAdd(pd,     w * v.x);
    atomicAdd(pd + 1, w * v.y);
}

// ---------------------------------------------------------------------------
// Fused dense epilogue of a TAGConv layer:
//   out = leaky_relu( x@W0 + h1@W1 + h2@W2 + h3@W3 + b + x )
// Exact-fp32 WMMA: V_WMMA_F32_16X16X4_F32, D(16x16) += A(16x4) x B(4x16).
// Block = 256 threads = 8 wave32s; each wave owns a 16-row x 64-col strip.
// All 4 weight matrices (4*64*64 f32 = exactly 64KB) staged in LDS once per
// block in PAIR-INTERLEAVED layout wp[t][c/2][n][2] so each B fragment
// (B[c+2hi][col], B[c+2hi+1][col]) is one aligned 8-byte ds_load_b64
// directly into an even VGPR pair (no shuffle movs).
// K is fused: 4 hops x 64 = 256, consumed 4 at a time -> 256 WMMAs per wave.
//
// Fragment mapping (ISA 7.12.2, 32-bit):
//   A: lane L (m=L&15, hi=L>>4): v0=A[m][2*hi], v1=A[m][2*hi+1]
//   B: v0=B[2*hi][m],  v1=B[2*hi+1][m]
//   C/D: acc[v] = element(row = v + 8*hi, col = m) of the 16x16 tile
// ---------------------------------------------------------------------------
__global__ __launch_bounds__(256) void tag_dense_kernel(
    const float* __restrict__ x,   // layer input == hop-0 features == residual
    const float* __restrict__ h1, const float* __restrict__ h2,
    const float* __restrict__ h3,
    const float* __restrict__ W,   // (4, 64, 64) row-major: W[k][c][n]
    const float* __restrict__ bias,
    float* __restrict__ out, int nrows)
{
    // wp[((t*32 + c/2)*64 + n)*2 + (c&1)] = W[t][c][n]   (64 KB total)
    __shared__ float wp[4 * 32 * FDIM * 2];

    // Stage weights: coalesced global read; LDS write addresses stride-2
    // floats across the wave -> 32 distinct banks, conflict-free.
    for (int idx = threadIdx.x; idx < 4 * FDIM * FDIM; idx += 256) {
        int t = idx >> 12;
        int c = (idx >> 6) & 63;
        int n = idx & 63;
        wp[(((t * 32 + (c >> 1)) * FDIM) + n) * 2 + (c & 1)] = W[idx];
    }
    __syncthreads();

    int wave = threadIdx.x >> 5;
    int lane = threadIdx.x & 31;
    int rowbase = (blockIdx.x * 8 + wave) * 16;
    if (rowbase >= nrows) return;          // uniform per-wave exit (EXEC all-1s for WMMA)

    int m  = lane & 15;
    int hi = lane >> 4;

    v8f acc[4] = {v8f{}, v8f{}, v8f{}, v8f{}};   // 4 N-tiles of 16 cols

    const float* hs[4] = {x, h1, h2, h3};

#pragma unroll
    for (int t = 0; t < 4; ++t) {
        // per-lane A base: row (rowbase+m), starting at feature 2*hi
        const float* ha = hs[t] + (size_t)(rowbase + m) * FDIM + 2 * hi;
        // per-lane B base in LDS: pair group (t*32 + hi), column m
        const float* wb = wp + (((size_t)t * 32 + hi) * FDIM + m) * 2;
#pragma unroll
        for (int c = 0; c < FDIM; c += 4) {
            v2f a = *(const v2f*)(ha + c);            // A[m][c+2hi], A[m][c+2hi+1]
            const float* wbc = wb + (c >> 1) * (FDIM * 2);
#pragma unroll
            for (int nt = 0; nt < 4; ++nt) {
                // one aligned 8B ds_load_b64: (W[t][c+2hi][col], W[t][c+2hi+1][col])
                v2f b = *(const v2f*)(wbc + nt * 32);
                acc[nt] = __builtin_amdgcn_wmma_f32_16x16x4_f32(
                    false, a, false, b, (short)0, acc[nt], false, false);
            }
        }
    }

    // Epilogue: bias + residual + leaky_relu(0.01), write 16x64 strip
#pragma unroll
    for (int nt = 0; nt < 4; ++nt) {
        int col = nt * 16 + m;
        float bb = bias[col];
#pragma unroll
        for (int v = 0; v < 8; ++v) {
            int row = rowbase + v + 8 * hi;
            float val = acc[nt][v] + bb + x[(size_t)row * FDIM + col];
            out[(size_t)row * FDIM + col] = val > 0.f ? val : 0.01f * val;
        }
    }
}

// ---------------------------------------------------------------------------
// Launch: per layer -> zero h1..h3, 3 propagate passes, fused WMMA dense.
// Layer 1 writes d_out (acts as y1); layer 2 reads d_out in-place safely
// (each element read+written by exactly one lane; row strips disjoint).
// Workspace: 3 * N * 64 * 4 bytes.
// ---------------------------------------------------------------------------
extern "C" void kernel_launch(void* const* d_in, const int* in_sizes, int n_in,
                              void* d_out, int out_size, void* d_ws, size_t ws_size,
                              hipStream_t stream)
{
    const float* y  = (const float*)d_in[0];
    const int*   ei = (const int*)d_in[1];
    const float* ew = (const float*)d_in[2];
    const float* W1 = (const float*)d_in[3];
    const float* b1 = (const float*)d_in[4];
    const float* W2 = (const float*)d_in[5];
    const float* b2 = (const float*)d_in[6];

    const int N = in_sizes[0] / FDIM;
    const int E = in_sizes[1] / 2;
    const int* src = ei;
    const int* dst = ei + E;

    float* h1 = (float*)d_ws;
    float* h2 = h1 + (size_t)N * FDIM;
    float* h3 = h2 + (size_t)N * FDIM;
    float* out = (float*)d_out;

    const long n4      = (long)3 * N * FDIM / 4;
    const int  zblocks = (int)((n4 + 255) / 256);
    const int  pblocks = (E + 7) / 8;
    const int  tiles   = (N + 15) / 16;
    const int  dblocks = (tiles + 7) / 8;

    // ---- layer 1 ----
    zero_kernel<<<zblocks, 256, 0, stream>>>((float4*)h1, n4);
    propagate_kernel<<<pblocks, 256, 0, stream>>>(y,  h1, src, dst, ew, E);
    propagate_kernel<<<pblocks, 256, 0, stream>>>(h1, h2, src, dst, ew, E);
    propagate_kernel<<<pblocks, 256, 0, stream>>>(h2, h3, src, dst, ew, E);
    tag_dense_kernel<<<dblocks, 256, 0, stream>>>(y, h1, h2, h3, W1, b1, out, N);

    // ---- layer 2 ----
    zero_kernel<<<zblocks, 256, 0, stream>>>((float4*)h1, n4);
    propagate_kernel<<<pblocks, 256, 0, stream>>>(out, h1, src, dst, ew, E);
    propagate_kernel<<<pblocks, 256, 0, stream>>>(h1,  h2, src, dst, ew, E);
    propagate_kernel<<<pblocks, 256, 0, stream>>>(h2,  h3, src, dst, ew, E);
    tag_dense_kernel<<<dblocks, 256, 0, stream>>>(out, h1, h2, h3, W2, b2, out, N);
}